// GCN_87076166959724
// MI455X (gfx1250) — compile-verified
//
#include <hip/hip_runtime.h>
#include <hip/hip_bf16.h>

typedef float v2f __attribute__((ext_vector_type(2)));
typedef float v8f __attribute__((ext_vector_type(8)));

// ---------------- degree / normalization ----------------
__global__ void k_deg_init(float* deg, int N) {
    int i = blockIdx.x * blockDim.x + threadIdx.x;
    if (i < N) deg[i] = 1.0f;   // self-loop contributes 1 to every node's degree
}

__global__ void k_deg_count(const int* __restrict__ cols, float* deg, long long E) {
    long long e = (long long)blockIdx.x * blockDim.x + threadIdx.x;
    if (e < E) atomicAdd(deg + cols[e], 1.0f);
}

__global__ void k_deg_rsqrt(float* deg, int N) {
    int i = blockIdx.x * blockDim.x + threadIdx.x;
    if (i < N) deg[i] = rsqrtf(deg[i]);   // deg >= 1 always
}

// ---------------- GEMM1: h1s = (X @ W1) * dis[row]  (128 -> 16) ----------------
__global__ void __launch_bounds__(256) k_gemm1(const float* __restrict__ x,
                                               const float* __restrict__ W1,
                                               const float* __restrict__ dis,
                                               float* __restrict__ h1s, int N) {
    __shared__ float sW[128 * 16];
    for (int i = threadIdx.x; i < 128 * 16; i += 256) sW[i] = W1[i];
    __syncthreads();

    const int wave = threadIdx.x >> 5;
    const int lane = threadIdx.x & 31;
    const int base = (blockIdx.x * 8 + wave) * 16;
    if (base >= N) return;                 // whole-wave exit: EXEC stays all-1 for WMMA

    const int l15 = lane & 15;             // A: M index, B/C: N index
    const int h   = lane >> 4;             // lane half
    const int rA  = (base + l15 < N) ? base + l15 : N - 1;
    const float* __restrict__ xr = x + (long long)rA * 128;

    v8f acc = {};
#pragma unroll
    for (int kk = 0; kk < 32; ++kk) {      // K = 128 in steps of 4
        const int k0 = kk * 4 + h * 2;
        v2f a; a.x = xr[k0];               a.y = xr[k0 + 1];
        v2f b; b.x = sW[k0 * 16 + l15];    b.y = sW[(k0 + 1) * 16 + l15];
        acc = __builtin_amdgcn_wmma_f32_16x16x4_f32(false, a, false, b,
                                                    (short)0, acc, false, false);
    }
#pragma unroll
    for (int i = 0; i < 8; ++i) {
        const int r = base + i + h * 8;    // C/D row mapping
        if (r < N) h1s[(long long)r * 16 + l15] = acc[i] * dis[r];
    }
}

// ---------------- GEMM2: gs = (H2 @ W2) * dis[row]  (16 -> 32) ----------------
__global__ void __launch_bounds__(256) k_gemm2(const float* __restrict__ h2s,
                                               const float* __restrict__ W2,
                                               const float* __restrict__ dis,
                                               float* __restrict__ gs, int N) {
    __shared__ float sW[16 * 32];
    for (int i = threadIdx.x; i < 16 * 32; i += 256) sW[i] = W2[i];
    __syncthreads();

    const int wave = threadIdx.x >> 5;
    const int lane = threadIdx.x & 31;
    const int base = (blockIdx.x * 8 + wave) * 16;
    if (base >= N) return;

    const int l15 = lane & 15;
    const int h   = lane >> 4;
    const int rA  = (base + l15 < N) ? base + l15 : N - 1;
    const float* __restrict__ ar = h2s + (long long)rA * 16;

    v8f acc0 = {}, acc1 = {};
#pragma unroll
    for (int kk = 0; kk < 4; ++kk) {       // K = 16 in steps of 4
        const int k0 = kk * 4 + h * 2;
        v2f a;  a.x  = ar[k0];                     a.y  = ar[k0 + 1];
        v2f b0; b0.x = sW[k0 * 32 + l15];          b0.y = sW[(k0 + 1) * 32 + l15];
        v2f b1; b1.x = sW[k0 * 32 + 16 + l15];     b1.y = sW[(k0 + 1) * 32 + 16 + l15];
        acc0 = __builtin_amdgcn_wmma_f32_16x16x4_f32(false, a, false, b0,
                                                     (short)0, acc0, false, false);
        acc1 = __builtin_amdgcn_wmma_f32_16x16x4_f32(false, a, false, b1,
                                                     (short)0, acc1, false, false);
    }
#pragma unroll
    for (int i = 0; i < 8; ++i) {
        const int r = base + i + h * 8;
        if (r < N) {
            const float d = dis[r];
            gs[(long long)r * 32 + l15]      = acc0[i] * d;
            gs[(long long)r * 32 + 16 + l15] = acc1[i] * d;
        }
    }
}

// ---------------- edge scatter: dst[col] += src[row] (F floats/edge) ----------------
template <int F>
__global__ void k_scatter(const int* __restrict__ rows, const int* __restrict__ cols,
                          const float* __restrict__ src, float* __restrict__ dst,
                          long long E) {
    constexpr int CH = F / 4;              // float4 chunks per edge
    long long tid = (long long)blockIdx.x * blockDim.x + threadIdx.x;
    long long e = tid / CH;
    int q = (int)(tid - e * CH);
    if (e >= E) return;
    const int r = rows[e], c = cols[e];
    const float4 v = ((const float4*)(src + (long long)r * F))[q];
    float* d = dst + (long long)c * F + q * 4;
    atomicAdd(d + 0, v.x);
    atomicAdd(d + 1, v.y);
    atomicAdd(d + 2, v.z);
    atomicAdd(d + 3, v.w);
}

// ---------------- layer-1 epilogue: h2s = relu(dis[n]*agg1 + b1) ----------------
__global__ void k_finish1(const float* __restrict__ agg1, const float* __restrict__ dis,
                          const float* __restrict__ b1, float* __restrict__ h2s, int N) {
    int i = blockIdx.x * blockDim.x + threadIdx.x;
    if (i >= N * 16) return;
    const int n = i >> 4, f = i & 15;
    const float v = dis[n] * agg1[i] + b1[f];
    h2s[i] = v > 0.0f ? v : 0.0f;
}

// ---------------- layer-2 epilogue: out = log_softmax(dis[n]*agg2 + b2) ----------------
__global__ void k_final(const float* __restrict__ agg2, const float* __restrict__ dis,
                        const float* __restrict__ b2, float* __restrict__ out, int N) {
    const int lane = threadIdx.x;          // 0..31 == one full wave == 32 features
    const int node = blockIdx.x * blockDim.y + threadIdx.y;
    if (node >= N) return;
    float z = dis[node] * agg2[(long long)node * 32 + lane] + b2[lane];
    float m = z;
#pragma unroll
    for (int k = 16; k >= 1; k >>= 1) m = fmaxf(m, __shfl_xor(m, k, 32));
    float s = expf(z - m);
#pragma unroll
    for (int k = 16; k >= 1; k >>= 1) s += __shfl_xor(s, k, 32);
    out[(long long)node * 32 + lane] = (z - m) - logf(s);
}

extern "C" void kernel_launch(void* const* d_in, const int* in_sizes, int n_in,
                              void* d_out, int out_size, void* d_ws, size_t ws_size,
                              hipStream_t stream) {
    const float* x  = (const float*)d_in[0];
    const int*   ei = (const int*)d_in[1];     // edge_index [2, E]
    const float* W1 = (const float*)d_in[2];
    const float* b1 = (const float*)d_in[3];
    const float* W2 = (const float*)d_in[4];
    const float* b2 = (const float*)d_in[5];
    float* out = (float*)d_out;

    const int       N = in_sizes[0] / 128;
    const long long E = (long long)in_sizes[1] / 2;
    const int* rows = ei;        // edge_index[0] (src)
    const int* cols = ei + E;    // edge_index[1] (dst)

    // workspace layout (floats)
    float* ws   = (float*)d_ws;
    float* dis  = ws;                               // N       (deg -> rsqrt in place)
    float* h1s  = dis  + N;                         // N*16
    float* agg1 = h1s  + (long long)N * 16;         // N*16
    float* h2s  = agg1 + (long long)N * 16;         // N*16
    float* gs   = h2s  + (long long)N * 16;         // N*32
    float* agg2 = gs   + (long long)N * 32;         // N*32

    const int nodeBlocks = (N + 255) / 256;
    const int edgeBlocks = (int)((E + 255) / 256);
    const int gemmBlocks = ((N + 15) / 16 + 7) / 8;     // 8 waves/block, 16 rows/wave

    // degree normalization
    k_deg_init <<<nodeBlocks, 256, 0, stream>>>(dis, N);
    k_deg_count<<<edgeBlocks, 256, 0, stream>>>(cols, dis, E);
    k_deg_rsqrt<<<nodeBlocks, 256, 0, stream>>>(dis, N);

    // layer 1
    k_gemm1<<<gemmBlocks, 256, 0, stream>>>(x, W1, dis, h1s, N);
    hipMemcpyAsync(agg1, h1s, (size_t)N * 16 * sizeof(float),
                   hipMemcpyDeviceToDevice, stream);      // self-loop term
    {
        long long t = E * 4;                              // 4 float4-chunks/edge
        k_scatter<16><<<(int)((t + 255) / 256), 256, 0, stream>>>(rows, cols, h1s, agg1, E);
    }
    k_finish1<<<(N * 16 + 255) / 256, 256, 0, stream>>>(agg1, dis, b1, h2s, N);

    // layer 2
    k_gemm2<<<gemmBlocks, 256, 0, stream>>>(h2s, W2, dis, gs, N);
    hipMemcpyAsync(agg2, gs, (size_t)N * 32 * sizeof(float),
                   hipMemcpyDeviceToDevice, stream);      // self-loop term
    {
        long long t = E * 8;                              // 8 float4-chunks/edge
        k_scatter<32><<<(int)((t + 255) / 256), 256, 0, stream>>>(rows, cols, gs, agg2, E);
    }
    dim3 fb(32, 8);
    k_final<<<(N + 7) / 8, fb, 0, stream>>>(agg2, dis, b2, out, N);
}